// TransformerEncoder_22617297780942
// MI455X (gfx1250) — compile-verified
//
#include <hip/hip_runtime.h>
#include <math.h>

// ---------------------------------------------------------------------------
// ViT encoder block, fp32, WMMA f32 16x16x4 path (memory-bound: stream each
// weight exactly once; M=25 padded to 32 handled inside one wave per N-strip).
// ---------------------------------------------------------------------------

typedef __attribute__((ext_vector_type(2))) float v2f;
typedef __attribute__((ext_vector_type(8))) float v8f;

#define EMB   4096
#define NTOK  25
#define MPAD  32
#define QKVW  12288
#define FFW   16384

// Epilogue modes
#define EPI_PE    0   // out = acc + bias[n] + (m<25 ? pos[m,n] : 0)
#define EPI_BIAS  1   // out = acc + bias[n]
#define EPI_RES   2   // C[m,n] += acc + bias[n]        (in-place residual)
#define EPI_GELU  3   // out = gelu(acc + bias[n])      (exact erf GELU)
#define EPI_OUT25 4   // m<25 only: out = aux[m,n] + acc + bias[n]

// ---------------------------------------------------------------------------
// im2col for Conv2d(C=256, k=4, s=4) on [1,256,20,20] -> A[32,4096], rows>=25 = 0
// patch token m = py*5+px ; element e = c*16 + kh*4 + kw  (matches pe_w layout)
// ---------------------------------------------------------------------------
__global__ __launch_bounds__(256) void im2col_kernel(const float* __restrict__ x,
                                                     float* __restrict__ A) {
    int i = blockIdx.x * 256 + threadIdx.x;          // 32*4096 = 131072
    if (i >= MPAD * EMB) return;
    int m = i >> 12, e = i & 4095;
    float v = 0.0f;
    if (m < NTOK) {
        int py = m / 5, px = m % 5;
        int c = e >> 4, kh = (e >> 2) & 3, kw = e & 3;
        v = x[c * 400 + (py * 4 + kh) * 20 + (px * 4 + kw)];
    }
    A[i] = v;
}

// ---------------------------------------------------------------------------
// Generic skinny GEMM: C[32,N] = A[32,K] @ B[K,N] (+ epilogue)
//   BT=false: B row-major [K,N]      BT=true: B stored as [N,K] (used for pe_w)
// One wave owns a 32(M) x 16(N) tile -> every B element loaded exactly once.
// 8 waves / block, block covers 128 columns.
// ---------------------------------------------------------------------------
template <bool BT, int EPI>
__global__ __launch_bounds__(256) void gemm_kernel(const float* __restrict__ A, int lda,
                                                   const float* __restrict__ B, int ldb,
                                                   int K,
                                                   float* __restrict__ C, int ldc,
                                                   const float* __restrict__ bias,
                                                   const float* __restrict__ aux) {
    const int lane = threadIdx.x & 31;
    const int wave = threadIdx.x >> 5;
    const int n0   = blockIdx.x * 128 + wave * 16;
    const int hlf  = lane >> 4;       // which K-pair (0 -> k0,k1 ; 1 -> k2,k3)
    const int l16  = lane & 15;       // M index (A frag) / N index (B frag)

    v8f acc0 = {};                    // C rows 0..15
    v8f acc1 = {};                    // C rows 16..31

    const float* a0 = A + l16 * lda + hlf * 2;
    const float* a1 = a0 + 16 * lda;

    if (!BT) {
        const float* bp = B + (hlf * 2) * ldb + n0 + l16;
#pragma unroll 4
        for (int k = 0; k < K; k += 4) {
            __builtin_prefetch(bp + (k + 64) * ldb, 0, 0);   // global_prefetch_b8
            v2f af0 = *(const v2f*)(a0 + k);
            v2f af1 = *(const v2f*)(a1 + k);
            v2f bf;
            bf.x = bp[k * ldb];
            bf.y = bp[(k + 1) * ldb];
            acc0 = __builtin_amdgcn_wmma_f32_16x16x4_f32(false, af0, false, bf,
                                                         (short)0, acc0, false, false);
            acc1 = __builtin_amdgcn_wmma_f32_16x16x4_f32(false, af1, false, bf,
                                                         (short)0, acc1, false, false);
        }
    } else {
        const float* bp = B + (n0 + l16) * ldb + hlf * 2;
#pragma unroll 4
        for (int k = 0; k < K; k += 4) {
            __builtin_prefetch(bp + k + 1024, 0, 0);
            v2f af0 = *(const v2f*)(a0 + k);
            v2f af1 = *(const v2f*)(a1 + k);
            v2f bf  = *(const v2f*)(bp + k);
            acc0 = __builtin_amdgcn_wmma_f32_16x16x4_f32(false, af0, false, bf,
                                                         (short)0, acc0, false, false);
            acc1 = __builtin_amdgcn_wmma_f32_16x16x4_f32(false, af1, false, bf,
                                                         (short)0, acc1, false, false);
        }
    }

    // Epilogue. C/D layout: VGPR r holds M = r + hlf*8 (acc0) / +16 (acc1), N = l16.
    const int n  = n0 + l16;
    const float bn = bias[n];
#pragma unroll
    for (int r = 0; r < 8; ++r) {
        int m0 = r + hlf * 8;
        int m1 = m0 + 16;
        float v0 = acc0[r] + bn;
        float v1 = acc1[r] + bn;
        if (EPI == EPI_PE) {
            if (m0 < NTOK) v0 += aux[m0 * EMB + n];
            if (m1 < NTOK) v1 += aux[m1 * EMB + n];
            C[m0 * ldc + n] = v0;
            C[m1 * ldc + n] = v1;
        } else if (EPI == EPI_BIAS) {
            C[m0 * ldc + n] = v0;
            C[m1 * ldc + n] = v1;
        } else if (EPI == EPI_RES) {
            C[m0 * ldc + n] += v0;
            C[m1 * ldc + n] += v1;
        } else if (EPI == EPI_GELU) {
            C[m0 * ldc + n] = 0.5f * v0 * (1.0f + erff(v0 * 0.70710678118654752f));
            C[m1 * ldc + n] = 0.5f * v1 * (1.0f + erff(v1 * 0.70710678118654752f));
        } else { // EPI_OUT25
            if (m0 < NTOK) C[m0 * ldc + n] = aux[m0 * EMB + n] + v0;
            if (m1 < NTOK) C[m1 * ldc + n] = aux[m1 * EMB + n] + v1;
        }
    }
}

// ---------------------------------------------------------------------------
// LayerNorm over last dim (4096) per row; biased variance (jnp.var default).
// ---------------------------------------------------------------------------
__global__ __launch_bounds__(256) void ln_kernel(const float* __restrict__ t,
                                                 const float* __restrict__ g,
                                                 const float* __restrict__ b,
                                                 float* __restrict__ h) {
    __shared__ float rs[256], rs2[256];
    const int row = blockIdx.x;
    const float* x = t + row * EMB;
    float s = 0.0f, s2 = 0.0f;
    for (int i = threadIdx.x; i < EMB; i += 256) {
        float v = x[i];
        s += v;
        s2 += v * v;
    }
    rs[threadIdx.x] = s;
    rs2[threadIdx.x] = s2;
    __syncthreads();
    for (int st = 128; st > 0; st >>= 1) {
        if (threadIdx.x < st) {
            rs[threadIdx.x]  += rs[threadIdx.x + st];
            rs2[threadIdx.x] += rs2[threadIdx.x + st];
        }
        __syncthreads();
    }
    const float mean = rs[0] * (1.0f / EMB);
    const float var  = rs2[0] * (1.0f / EMB) - mean * mean;
    const float inv  = rsqrtf(var + 1e-5f);
    for (int i = threadIdx.x; i < EMB; i += 256)
        h[row * EMB + i] = (x[i] - mean) * inv * g[i] + b[i];
}

// ---------------------------------------------------------------------------
// Attention: one block per head. qkv[tok, head*768 + d*3 + {0,1,2}] = q/k/v.
// energy = q@k^T, softmax per row, then /sqrt(EMB) AFTER softmax (faithful),
// then att@v -> o[tok, head*256 + d].
// ---------------------------------------------------------------------------
__global__ __launch_bounds__(256) void attn_kernel(const float* __restrict__ qkv,
                                                   float* __restrict__ o) {
    __shared__ float sQ[NTOK * 256];
    __shared__ float sKV[NTOK * 256];
    __shared__ float sE[NTOK * 26];
    const int head = blockIdx.x;
    const int tid  = threadIdx.x;
    const int base = head * 768;

    for (int i = tid; i < NTOK * 256; i += 256) {
        int tok = i >> 8, d = i & 255;
        const float* p = qkv + tok * QKVW + base + d * 3;
        sQ[i]  = p[0];
        sKV[i] = p[1];
    }
    __syncthreads();

    for (int i = tid; i < NTOK * NTOK; i += 256) {
        int q = i / NTOK, kk = i % NTOK;
        float s = 0.0f;
        for (int d = 0; d < 256; ++d) s += sQ[q * 256 + d] * sKV[kk * 256 + d];
        sE[q * 26 + kk] = s;
    }
    __syncthreads();

    if (tid < NTOK) {
        float mx = -INFINITY;
        for (int kk = 0; kk < NTOK; ++kk) mx = fmaxf(mx, sE[tid * 26 + kk]);
        float sum = 0.0f;
        for (int kk = 0; kk < NTOK; ++kk) {
            float e = __expf(sE[tid * 26 + kk] - mx);
            sE[tid * 26 + kk] = e;
            sum += e;
        }
        float inv = 1.0f / (sum * 64.0f);   // softmax then / sqrt(4096)
        for (int kk = 0; kk < NTOK; ++kk) sE[tid * 26 + kk] *= inv;
    }
    __syncthreads();

    for (int i = tid; i < NTOK * 256; i += 256) {   // reload with V
        int tok = i >> 8, d = i & 255;
        sKV[i] = qkv[tok * QKVW + base + d * 3 + 2];
    }
    __syncthreads();

    for (int i = tid; i < NTOK * 256; i += 256) {
        int q = i >> 8, d = i & 255;
        float s = 0.0f;
#pragma unroll
        for (int kk = 0; kk < NTOK; ++kk) s += sE[q * 26 + kk] * sKV[kk * 256 + d];
        o[q * EMB + head * 256 + d] = s;
    }
}

// ---------------------------------------------------------------------------
extern "C" void kernel_launch(void* const* d_in, const int* in_sizes, int n_in,
                              void* d_out, int out_size, void* d_ws, size_t ws_size,
                              hipStream_t stream) {
    const float* x      = (const float*)d_in[0];
    const float* pe_w   = (const float*)d_in[1];
    const float* pe_b   = (const float*)d_in[2];
    const float* pos    = (const float*)d_in[3];
    const float* ln_g   = (const float*)d_in[4];
    const float* ln_b   = (const float*)d_in[5];
    const float* qkv_w  = (const float*)d_in[6];
    const float* qkv_b  = (const float*)d_in[7];
    const float* proj_w = (const float*)d_in[8];
    const float* proj_b = (const float*)d_in[9];
    const float* ff1_w  = (const float*)d_in[10];
    const float* ff1_b  = (const float*)d_in[11];
    const float* ff2_w  = (const float*)d_in[12];
    const float* ff2_b  = (const float*)d_in[13];
    float* out = (float*)d_out;

    float* ws   = (float*)d_ws;
    float* A_pe = ws;                       // 32*4096
    float* t    = A_pe + MPAD * EMB;        // 32*4096 (token embeddings / residual)
    float* h    = t    + MPAD * EMB;        // 32*4096 (LN output)
    float* qkv  = h    + MPAD * EMB;        // 32*12288
    float* o    = qkv  + MPAD * QKVW;       // 32*4096 (attn output)
    float* ffh  = o    + MPAD * EMB;        // 32*16384 (FF hidden)

    // 1) im2col
    im2col_kernel<<<(MPAD * EMB) / 256, 256, 0, stream>>>(x, A_pe);
    // 2) patch-embed GEMM: t = A_pe @ pe_w^T + pe_b + pos   (pe_w is [4096out,4096in])
    gemm_kernel<true, EPI_PE><<<EMB / 128, 256, 0, stream>>>(
        A_pe, EMB, pe_w, EMB, EMB, t, EMB, pe_b, pos);
    // 3) LN
    ln_kernel<<<MPAD, 256, 0, stream>>>(t, ln_g, ln_b, h);
    // 4) QKV GEMM: qkv = h @ qkv_w + qkv_b
    gemm_kernel<false, EPI_BIAS><<<QKVW / 128, 256, 0, stream>>>(
        h, EMB, qkv_w, QKVW, EMB, qkv, QKVW, qkv_b, nullptr);
    // 5) attention (one block per head)
    attn_kernel<<<16, 256, 0, stream>>>(qkv, o);
    // 6) proj GEMM with residual: t += o @ proj_w + proj_b
    gemm_kernel<false, EPI_RES><<<EMB / 128, 256, 0, stream>>>(
        o, EMB, proj_w, EMB, EMB, t, EMB, proj_b, nullptr);
    // 7) LN (shared params)
    ln_kernel<<<MPAD, 256, 0, stream>>>(t, ln_g, ln_b, h);
    // 8) FF1 GEMM + exact GELU
    gemm_kernel<false, EPI_GELU><<<FFW / 128, 256, 0, stream>>>(
        h, EMB, ff1_w, FFW, EMB, ffh, FFW, ff1_b, nullptr);
    // 9) FF2 GEMM + residual, straight into d_out (flat [25,4096] == [1,256,20,20])
    gemm_kernel<false, EPI_OUT25><<<EMB / 128, 256, 0, stream>>>(
        ffh, FFW, ff2_w, EMB, FFW, out, EMB, ff2_b, t);
}